// SSGC_15934328669027
// MI455X (gfx1250) — compile-verified
//
#include <hip/hip_runtime.h>

#define FD 64
#define ALPHA_F 0.6f
#define COEF_F 0.05f   // (1-ALPHA)/K = 0.4/8

typedef __attribute__((ext_vector_type(2))) float v2f;
typedef __attribute__((ext_vector_type(8))) float v8f;

// ---------------- degree / normalization ----------------
__global__ void k_init_deg(float* __restrict__ deg, int n) {
  int i = blockIdx.x * blockDim.x + threadIdx.x;
  if (i < n) deg[i] = 1.0f;            // self loop contributes 1
}

__global__ void k_scatter_deg(const int* __restrict__ dst, float* __restrict__ deg, int e) {
  int i = blockIdx.x * blockDim.x + threadIdx.x;
  if (i < e) atomicAdd(&deg[dst[i]], 1.0f);
}

__global__ void k_dinv(float* __restrict__ deg, int n) {
  int i = blockIdx.x * blockDim.x + threadIdx.x;
  if (i < n) deg[i] = rsqrtf(deg[i]);  // deg >= 1 always
}

// ---------------- propagation ----------------
__global__ void k_init_h(const float* __restrict__ x, float* __restrict__ h, int total) {
  int i = blockIdx.x * blockDim.x + threadIdx.x;
  if (i < total) h[i] = ALPHA_F * x[i];
}

// xnew[i,:] = dinv[i]^2 * xk[i,:]   (self-loop term, also zero-initializes xnew)
__global__ void k_self(const float* __restrict__ xk, const float* __restrict__ dinv,
                       float* __restrict__ xnew, int total) {
  int i = blockIdx.x * blockDim.x + threadIdx.x;
  if (i < total) {
    float s = dinv[i >> 6];
    xnew[i] = s * s * xk[i];
  }
}

// one edge per 64 lanes: coalesced 256B row gather, fp32 atomic scatter (L2-resident)
__global__ void k_scatter(const int* __restrict__ src, const int* __restrict__ dst,
                          const float* __restrict__ dinv, const float* __restrict__ xk,
                          float* __restrict__ xnew, int e) {
  int gid = blockIdx.x * blockDim.x + threadIdx.x;
  int ed = gid >> 6;
  int f  = gid & 63;
  if (ed < e) {
    int s = src[ed];
    int d = dst[ed];
    float nrm = dinv[s] * dinv[d];
    atomicAdd(&xnew[(d << 6) + f], nrm * xk[(s << 6) + f]);
  }
}

__global__ void k_accum(const float* __restrict__ xnew, float* __restrict__ h, int total) {
  int i = blockIdx.x * blockDim.x + threadIdx.x;
  if (i < total) h[i] += COEF_F * xnew[i];
}

// ---------------- fused 3x GEMM(64x64) MLP with fp32 WMMA ----------------
// One wave computes 16 rows of the output. Three chained GEMMs:
//   t1 = relu(h @ Wc^T + bc); t2 = relu(t1 @ W1^T + b1); out = t2 @ W2^T + b2
// C/D results are round-tripped through a per-wave LDS tile to re-load in
// A-fragment layout for the next GEMM.
__global__ __launch_bounds__(256) void k_mlp_wmma(
    const float* __restrict__ h,
    const float* __restrict__ wc, const float* __restrict__ bc,
    const float* __restrict__ w1, const float* __restrict__ b1,
    const float* __restrict__ w2, const float* __restrict__ b2,
    float* __restrict__ out, int n) {
  constexpr int LDW = 68;                 // stride pad: bank-conflict-free column walks
  __shared__ float sW[64 * LDW];          // current weight matrix (row-major, padded)
  __shared__ float sT[8 * 16 * LDW];      // per-wave 16x64 staging tiles

  const int tid  = threadIdx.x;
  const int lane = tid & 31;
  const int wv   = tid >> 5;              // wave id 0..7 (wave32)
  const int cn   = lane & 15;             // N-within-tile / M-row selector
  const int hi   = lane >> 4;             // 0: lanes 0-15, 1: lanes 16-31
  const int m0   = (blockIdx.x * 8 + wv) * 16;
  float* st = &sT[wv * 16 * LDW];

  auto loadW = [&](const float* __restrict__ w) {
    for (int idx = tid; idx < 64 * 64; idx += 256)
      sW[(idx >> 6) * LDW + (idx & 63)] = w[idx];
  };

  v8f a0, a1, a2, a3;

  // A-fragment K-base for this lane at K-step k0: lanes 0-15 hold K=k0,k0+1; 16-31 hold k0+2,k0+3
  auto kbase = [&](int k0) { return k0 + (hi << 1); };

  // one GEMM over K=64 given a per-lane A-row pointer (stride-1 in K)
  auto gemm = [&](const float* __restrict__ arow, int astride) {
    a0 = v8f{}; a1 = v8f{}; a2 = v8f{}; a3 = v8f{};
    for (int k0 = 0; k0 < 64; k0 += 4) {
      int kb = kbase(k0);
      v2f a;
      a.x = arow[kb];
      a.y = arow[kb + 1];
      const float* wb = &sW[cn * LDW + kb];   // row n0+cn of W, cols kb,kb+1
      v2f b;
      b.x = wb[0];               b.y = wb[1];
      a0 = __builtin_amdgcn_wmma_f32_16x16x4_f32(false, a, false, b, (short)0, a0, false, false);
      b.x = wb[16 * LDW];        b.y = wb[16 * LDW + 1];
      a1 = __builtin_amdgcn_wmma_f32_16x16x4_f32(false, a, false, b, (short)0, a1, false, false);
      b.x = wb[32 * LDW];        b.y = wb[32 * LDW + 1];
      a2 = __builtin_amdgcn_wmma_f32_16x16x4_f32(false, a, false, b, (short)0, a2, false, false);
      b.x = wb[48 * LDW];        b.y = wb[48 * LDW + 1];
      a3 = __builtin_amdgcn_wmma_f32_16x16x4_f32(false, a, false, b, (short)0, a3, false, false);
    }
    (void)astride;
  };

  // C/D layout -> staging tile, with bias + relu
  auto stage_relu = [&](const float* __restrict__ bias) {
    for (int r = 0; r < 8; ++r) {
      int rr = (hi << 3) + r;               // M = r (lanes 0-15) or 8+r (lanes 16-31)
      float* srow = &st[rr * LDW];
      srow[ 0 + cn] = fmaxf(a0[r] + bias[ 0 + cn], 0.0f);
      srow[16 + cn] = fmaxf(a1[r] + bias[16 + cn], 0.0f);
      srow[32 + cn] = fmaxf(a2[r] + bias[32 + cn], 0.0f);
      srow[48 + cn] = fmaxf(a3[r] + bias[48 + cn], 0.0f);
    }
  };

  // ---- GEMM 1: A from global h (clamped row for tail; EXEC stays all-ones) ----
  loadW(wc);
  __syncthreads();
  {
    int row = m0 + cn;
    if (row >= n) row = n - 1;
    gemm(h + (size_t)row * FD, FD);
  }
  stage_relu(bc);
  __syncthreads();

  // ---- GEMM 2: A from staging tile ----
  loadW(w1);
  __syncthreads();
  gemm(&st[cn * LDW], LDW);
  stage_relu(b1);
  __syncthreads();

  // ---- GEMM 3: A from staging tile, write out with bias (no relu) ----
  loadW(w2);
  __syncthreads();
  gemm(&st[cn * LDW], LDW);

  for (int r = 0; r < 8; ++r) {
    int row = m0 + (hi << 3) + r;
    if (row < n) {
      float* orow = out + (size_t)row * FD;
      orow[ 0 + cn] = a0[r] + b2[ 0 + cn];
      orow[16 + cn] = a1[r] + b2[16 + cn];
      orow[32 + cn] = a2[r] + b2[32 + cn];
      orow[48 + cn] = a3[r] + b2[48 + cn];
    }
  }
}

extern "C" void kernel_launch(void* const* d_in, const int* in_sizes, int n_in,
                              void* d_out, int out_size, void* d_ws, size_t ws_size,
                              hipStream_t stream) {
  const float* x  = (const float*)d_in[0];
  const float* wc = (const float*)d_in[1];
  const float* bc = (const float*)d_in[2];
  const float* w1 = (const float*)d_in[3];
  const float* b1 = (const float*)d_in[4];
  const float* w2 = (const float*)d_in[5];
  const float* b2 = (const float*)d_in[6];
  const int*   ei = (const int*)d_in[7];   // JAX default: int64 request -> int32 arrays

  const int N = in_sizes[0] / FD;
  const int E = in_sizes[7] / 2;
  const int* src = ei;
  const int* dst = ei + E;
  float* out = (float*)d_out;

  float* ws   = (float*)d_ws;
  size_t npad = ((size_t)N + 63) & ~(size_t)63;
  float* dinv = ws;
  float* bufA = dinv + npad;
  float* bufB = bufA + (size_t)N * FD;
  float* hbuf = bufB + (size_t)N * FD;

  const int T = 256;
  const int total = N * FD;
  const int ngrid = (N + T - 1) / T;
  const int tgrid = (total + T - 1) / T;
  const int egrid = (E + T - 1) / T;
  const int sgrid = (int)(((long long)E * FD + T - 1) / T);

  // gcn_norm: deg (with self loop) -> dinv
  k_init_deg<<<ngrid, T, 0, stream>>>(dinv, N);
  k_scatter_deg<<<egrid, T, 0, stream>>>(dst, dinv, E);
  k_dinv<<<ngrid, T, 0, stream>>>(dinv, N);

  // h = alpha * x
  k_init_h<<<tgrid, T, 0, stream>>>(x, hbuf, total);

  // K=8 propagation hops, ping-pong
  const float* cur = x;
  float* bufs[2] = {bufA, bufB};
  for (int k = 0; k < 8; ++k) {
    float* nxt = bufs[k & 1];
    k_self<<<tgrid, T, 0, stream>>>(cur, dinv, nxt, total);
    k_scatter<<<sgrid, T, 0, stream>>>(src, dst, dinv, cur, nxt, E);
    k_accum<<<tgrid, T, 0, stream>>>(nxt, hbuf, total);
    cur = nxt;
  }

  // fused MLP with fp32 WMMA: 8 waves/block * 16 rows = 128 rows per block
  const int mgrid = (N + 127) / 128;
  k_mlp_wmma<<<mgrid, T, 0, stream>>>(hbuf, wc, bc, w1, b1, w2, b2, out, N);
}